// Embedding_24807731102389
// MI455X (gfx1250) — compile-verified
//
#include <hip/hip_runtime.h>

typedef __attribute__((ext_vector_type(2))) float v2f;
typedef __attribute__((ext_vector_type(8))) float v8f;

#define BATCH 8192
#define NCAT  20
#define FEAT  200
#define HID   100
#define DIM   64

// ---------------------------------------------------------------------------
// Kernel 1: categorical embedding gather.  [B,20] int32 -> [B,20,64] f32.
// One float4 per thread; 16 consecutive threads stream one 256B embedding row.
// ---------------------------------------------------------------------------
__global__ __launch_bounds__(256) void cat_gather_kernel(
    const int* __restrict__ xcat,      // [B*20]
    const float* __restrict__ emb,     // [10000, 64]
    float* __restrict__ out)           // [B*20*64]
{
    int t = blockIdx.x * blockDim.x + threadIdx.x;
    const int total = BATCH * NCAT * (DIM / 4);
    if (t >= total) return;
    int within = t & 15;               // float4 index within the 64-f32 row
    int row    = t >> 4;               // b*20 + cat
    int tok    = xcat[row];
    const float4* src = (const float4*)emb + (size_t)tok * (DIM / 4) + within;
    ((float4*)out)[t] = *src;
}

// ---------------------------------------------------------------------------
// Kernel 2: per-feature MLP with f32 WMMA.
//   h[m,k] = sigmoid(x[m,f]*W1[f,k] + b1[f,k])          (16 x 100, in LDS)
//   C[m,n] = sum_k h[m,k]*W2[f,k,n] + b2[f,n]           (16 x 64 per block)
// Block = 128 threads = 4 waves; wave w computes the 16x16 N-tile at n0=w*16
// via 25 x V_WMMA_F32_16X16X4_F32.
// ---------------------------------------------------------------------------
__global__ __launch_bounds__(128) void cont_mlp_wmma_kernel(
    const float* __restrict__ xc,      // [B, 200]  (flatten of [B,10,20])
    const float* __restrict__ W1,      // [200, 100]
    const float* __restrict__ b1,      // [200, 100]
    const float* __restrict__ W2,      // [200, 100, 64]
    const float* __restrict__ b2,      // [200, 64]
    float* __restrict__ out)           // [B, 200, 64]
{
    __shared__ float hbuf[16 * HID];   // 6.4 KB; row stride 100 -> conflict-free

    const int f   = blockIdx.y;
    const int m0  = blockIdx.x * 16;
    const int tid = threadIdx.x;

    // ---- Phase 1: cooperative sigmoid MLP layer 1 into LDS -----------------
    const float* W1f = W1 + f * HID;
    const float* b1f = b1 + f * HID;
    for (int idx = tid; idx < 16 * HID; idx += 128) {
        int m = idx / HID;
        int k = idx - m * HID;
        float x = xc[(size_t)(m0 + m) * FEAT + f];
        float t = fmaf(x, W1f[k], b1f[k]);
        hbuf[idx] = 1.0f / (1.0f + __expf(-t));   // v_exp_f32 + v_rcp_f32
    }
    __syncthreads();

    // ---- Phase 2: 16x16x100 GEMM tile per wave -----------------------------
    const int wave = tid >> 5;
    const int lane = tid & 31;
    const int half = lane >> 4;        // 0: lanes 0-15, 1: lanes 16-31
    const int nl   = lane & 15;
    const int n0   = wave * 16;
    const int col  = n0 + nl;

    const float* W2f = W2 + (size_t)f * HID * DIM;

    v8f c = {0.f, 0.f, 0.f, 0.f, 0.f, 0.f, 0.f, 0.f};

#pragma unroll
    for (int k0 = 0; k0 < HID; k0 += 4) {
        // A fragment (16x4 f32): lane holds M=nl, K = k0 + 2*half + {0,1}.
        int kA = k0 + (half << 1);
        v2f a = *(const v2f*)&hbuf[nl * HID + kA];   // ds_load_b64, 8B aligned

        // B fragment (4x16 f32): v0 = W2[kA][col], v1 = W2[kA+1][col].
        const float* bp = W2f + (size_t)kA * DIM + col;
        v2f bvec;
        bvec.x = bp[0];
        bvec.y = bp[DIM];

        c = __builtin_amdgcn_wmma_f32_16x16x4_f32(
                /*neg_a=*/false, a, /*neg_b=*/false, bvec,
                /*c_mod=*/(short)0, c, /*reuse_a=*/false, /*reuse_b=*/false);
    }

    // ---- Epilogue: bias + coalesced store ----------------------------------
    float bias = b2[f * DIM + col];
    // C layout: VGPR i -> row M = i + 8*half, col = n0 + nl.
    float* op = out + ((size_t)(m0 + half * 8) * FEAT + f) * DIM + col;
    const size_t rstride = (size_t)FEAT * DIM;   // one B-row of cont output
#pragma unroll
    for (int i = 0; i < 8; ++i) {
        op[(size_t)i * rstride] = c[i] + bias;
    }
}

// ---------------------------------------------------------------------------
extern "C" void kernel_launch(void* const* d_in, const int* in_sizes, int n_in,
                              void* d_out, int out_size, void* d_ws, size_t ws_size,
                              hipStream_t stream) {
    const int*   xcat  = (const int*)d_in[0];     // [8192, 20] int32
    const float* xcont = (const float*)d_in[1];   // [8192, 10, 20] f32
    const float* emb   = (const float*)d_in[2];   // [10000, 64] f32
    const float* W1    = (const float*)d_in[3];   // [200, 100] f32
    const float* b1    = (const float*)d_in[4];   // [200, 100] f32
    const float* W2    = (const float*)d_in[5];   // [200, 100, 64] f32
    const float* b2    = (const float*)d_in[6];   // [200, 64] f32

    float* cat_out  = (float*)d_out;                              // [B,20,64]
    float* cont_out = cat_out + (size_t)BATCH * NCAT * DIM;       // [B,200,64]

    {
        const int total  = BATCH * NCAT * (DIM / 4);
        const int blocks = (total + 255) / 256;
        cat_gather_kernel<<<blocks, 256, 0, stream>>>(xcat, emb, cat_out);
    }
    {
        dim3 grid(BATCH / 16, FEAT);   // (512, 200)
        cont_mlp_wmma_kernel<<<grid, 128, 0, stream>>>(xcont, W1, b1, W2, b2,
                                                       cont_out);
    }
}